// ProbAttention_84567906058561
// MI455X (gfx1250) — compile-verified
//
#include <hip/hip_runtime.h>
#include <math.h>

// ProbAttention for MI455X (gfx1250, wave32, WMMA).
// B=8, L=1024, H=8, E=64, 32 patches of 64, prune-rate 0.1.
//
// Strategy: one workgroup per (b, h, 32-row tile). The full 32x1024 fp32 score
// row-block lives in LDS (~132 KB of the WGP's 320 KB) because the exact
// per-row quantile needs every element of the row before softmax. Both GEMMs
// use V_WMMA_F32_16X16X4_F32 to stay bit-faithful to the fp32 reference
// (quantile pruning is discontinuous in the scores, so low precision is risky).

typedef __attribute__((ext_vector_type(2))) float v2f;
typedef __attribute__((ext_vector_type(8))) float v8f;

#define LDS_STRIDE 1028   // 1024 + 4 dword pad -> conflict-free column access

__device__ __forceinline__ int wredSum(int v) {
#pragma unroll
  for (int m = 16; m; m >>= 1) v += __shfl_xor(v, m, 32);
  return v;
}
__device__ __forceinline__ float wredSumF(float v) {
#pragma unroll
  for (int m = 16; m; m >>= 1) v += __shfl_xor(v, m, 32);
  return v;
}
__device__ __forceinline__ float wredMaxF(float v) {
#pragma unroll
  for (int m = 16; m; m >>= 1) v = fmaxf(v, __shfl_xor(v, m, 32));
  return v;
}

// k-th smallest (0-indexed) of the 1024 uint keys held 32-per-lane across the
// wave. Radix bit descent: build max v with count(keys < v) <= k; that v is
// exactly the k-th order statistic. All lanes converge to the same result.
__device__ __forceinline__ unsigned kthSmallest(const unsigned (&key)[32], int k) {
  unsigned res = 0u;
  for (int bit = 31; bit >= 0; --bit) {
    unsigned cand = res | (1u << bit);
    int cnt = 0;
#pragma unroll
    for (int j = 0; j < 32; ++j) cnt += (key[j] < cand) ? 1 : 0;
    cnt = wredSum(cnt);
    if (cnt <= k) res = cand;
  }
  return res;
}

__global__ __launch_bounds__(256)
void prob_attn_kernel(const float* __restrict__ Q, const float* __restrict__ K,
                      const float* __restrict__ V, const int* __restrict__ starts,
                      float* __restrict__ O) {
  constexpr int L = 1024, H = 8, E = 64, RT = 32;  // RT = rows per block

  __shared__ float S[RT * LDS_STRIDE];   // score tile -> probability tile (in place)
  __shared__ unsigned inpatch[L];        // bit p set iff position is inside patch p
  __shared__ float inv_sum[RT];

  const int tid  = threadIdx.x;
  const int wave = tid >> 5;
  const int lane = tid & 31;
  const int lh   = lane & 15;
  const int klo  = (lane < 16) ? 0 : 2;  // K-slot base for this half-wave (A/B frags)
  const int hih  = (lane < 16) ? 0 : 8;  // C/D row offset for upper half-wave
  const int b = blockIdx.z, h = blockIdx.y;
  const int row_base = blockIdx.x * RT;

  // ---- patch-membership bitmask (mask[i][j] = (inpatch[i] & inpatch[j]) != 0)
  for (int i = tid; i < L; i += 256) {
    unsigned m = 0u;
#pragma unroll
    for (int p = 0; p < 32; ++p) {
      const int s0 = starts[p];
      if (i >= s0 && i < s0 + 64) m |= (1u << p);
    }
    inpatch[i] = m;
  }
  __syncthreads();

  // ---- Phase 1: S = mask ? Q K^T : 0   (fp32 WMMA 16x16x4, E-loop of 16)
  const int mt = wave & 1;       // which 16-row half of the 32-row tile
  const int r0 = mt * 16;
  {
    // A fragment (16x4 fp32): lane<16 holds K={0,1}, lane>=16 holds K={2,3};
    // cache all 16 E-chunks in registers, reused across this wave's 16 column tiles.
    const float* qrow = Q + ((size_t)(b * L + row_base + r0 + lh) * H + h) * E;
    v2f afrag[16];
#pragma unroll
    for (int c = 0; c < 16; ++c)
      afrag[c] = *(const v2f*)(qrow + 4 * c + klo);

    unsigned prow[8];
#pragma unroll
    for (int i = 0; i < 8; ++i)
      prow[i] = inpatch[row_base + r0 + i + hih];

    for (int nt = (wave >> 1); nt < 64; nt += 4) {
      const int n0 = nt * 16;
      const float* krow = K + ((size_t)(b * L + n0 + lh) * H + h) * E;
      v8f acc = {};
#pragma unroll
      for (int c = 0; c < 16; ++c) {
        const v2f bfrag = *(const v2f*)(krow + 4 * c + klo);  // B = K^T 4x16 slab
        acc = __builtin_amdgcn_wmma_f32_16x16x4_f32(
            false, afrag[c], false, bfrag, (short)0, acc, false, false);
      }
      const unsigned pc = inpatch[n0 + lh];
#pragma unroll
      for (int i = 0; i < 8; ++i) {
        const float val = (prow[i] & pc) ? acc[i] : 0.0f;
        S[(r0 + i + hih) * LDS_STRIDE + n0 + lh] = val;
      }
    }
  }
  __syncthreads();

  // ---- Phase 2: exact 0.1-quantile threshold, prune, exp (softmax numerators)
  // q*(n-1) = 0.1*1023 = 102.3 -> interpolate between sorted |s|[102] and [103].
  for (int rr = 0; rr < 4; ++rr) {
    const int rl = wave * 4 + rr;
    float* srow = &S[rl * LDS_STRIDE];
    float sreg[32];
    unsigned kreg[32];
#pragma unroll
    for (int j = 0; j < 32; ++j) {
      const float s = srow[lane + 32 * j];
      sreg[j] = s;
      kreg[j] = __float_as_uint(s) & 0x7fffffffu;  // |s| bits; uint order == float order
    }
    const float a102 = __uint_as_float(kthSmallest(kreg, 102));
    const float a103 = __uint_as_float(kthSmallest(kreg, 103));
    const float thr  = a102 + 0.3f * (a103 - a102);
    const unsigned tb = __float_as_uint(thr);      // |s| >= thr  <=>  kreg >= tb

    float mx = -INFINITY;
#pragma unroll
    for (int j = 0; j < 32; ++j)
      if (kreg[j] >= tb) mx = fmaxf(mx, sreg[j]);
    mx = wredMaxF(mx);

    float sum = 0.0f;
#pragma unroll
    for (int j = 0; j < 32; ++j) {
      const float p = (kreg[j] >= tb) ? expf(sreg[j] - mx) : 0.0f;
      sum += p;
      srow[lane + 32 * j] = p;                     // overwrite scores with numerators
    }
    sum = wredSumF(sum);
    if (lane == 0) inv_sum[rl] = 1.0f / sum;
  }
  __syncthreads();

  // ---- Phase 3: O = (P / rowsum) * V   (one 16x16 output tile per wave,
  //      K-loop 1024/4 fp32 WMMAs; A frags straight from LDS, conflict-free)
  {
    const int mt2 = wave >> 2;
    const int n0  = (wave & 3) * 16;
    const float* vcol = V + ((size_t)(b * L) * H + h) * E + n0 + lh; // V[s][e=n0+lh]
    const float* arow = &S[(mt2 * 16 + lh) * LDS_STRIDE];
    v8f acc = {};
#pragma unroll 4
    for (int kk = 0; kk < L; kk += 4) {
      const v2f af = *(const v2f*)(arow + kk + klo);
      v2f bf;
      bf.x = vcol[(size_t)(kk + klo) * (H * E)];
      bf.y = vcol[(size_t)(kk + klo + 1) * (H * E)];
      acc = __builtin_amdgcn_wmma_f32_16x16x4_f32(
          false, af, false, bf, (short)0, acc, false, false);
    }
#pragma unroll
    for (int i = 0; i < 8; ++i) {
      const int rl = mt2 * 16 + i + hih;
      const int l  = row_base + rl;
      O[((size_t)(b * L + l) * H + h) * E + n0 + lh] = acc[i] * inv_sum[rl];
    }
  }
}

extern "C" void kernel_launch(void* const* d_in, const int* in_sizes, int n_in,
                              void* d_out, int out_size, void* d_ws, size_t ws_size,
                              hipStream_t stream) {
  const float* Q = (const float*)d_in[0];
  const float* K = (const float*)d_in[1];
  const float* V = (const float*)d_in[2];
  const int* starts = (const int*)d_in[3];
  float* O = (float*)d_out;

  dim3 grid(1024 / 32, 8, 8);  // (row tiles, H, B)
  dim3 block(256);             // 8 wave32 waves on one WGP
  hipLaunchKernelGGL(prob_attn_kernel, grid, block, 0, stream, Q, K, V, starts, O);
}